// BoundaryLoss_66030827209007
// MI455X (gfx1250) — compile-verified
//
#include <hip/hip_runtime.h>
#include <hip/hip_bf16.h>
#include <math.h>

// Boundary loss for pred[16,4,256,256] f32, target[16,256,256] i32 -> scalar f32.
// Per class c in 1..3 (buffers reused, stream-ordered):
//   rowEDT:     exact row min-plus pass (bit-matches reference fp32 semantics)
//   colEDTLoss: TDM-staged column min-plus pass + sqrt + softmax + |p-tc|*dist
//   finalize:   deterministic fixed-order global reduction -> d_out[0]
// CDNA5 paths: v_wmma_f32_16x16x4_f32 wave reductions, tensor_load_to_lds +
// s_wait_tensorcnt LDS staging, wave32 ds_bpermute shuffle.

typedef __attribute__((ext_vector_type(2))) float v2f;
typedef __attribute__((ext_vector_type(8))) float v8f;
typedef __attribute__((ext_vector_type(4))) unsigned int v4u;
typedef __attribute__((ext_vector_type(4))) int v4i;
typedef __attribute__((ext_vector_type(8))) int v8i;

#define HH 256
#define WW 256
#define NB 16
#define NCLS 3
#define IMG (HH * WW)            // 65536
#define IT 4                     // output rows per block in colEDTLoss
#define CH 16                    // staged f2 rows per chunk
#define NBLK2 (NB * (HH / IT))   // 1024 blocks per class in colEDTLoss
#define NSUMS (NCLS * NBLK2)     // 3072 block sums total

#if __has_builtin(__builtin_amdgcn_tensor_load_to_lds) && \
    __has_builtin(__builtin_amdgcn_s_wait_tensorcnt)
#define HAVE_TDM 1
#else
#define HAVE_TDM 0
#endif

// Full wave32 sum via one f32 WMMA: A(16x4) holds the 32 lane values
// (lanes 0-15 -> A[m][0], lanes 16-31 -> A[m][2]), B = ones(4x16).
// Summing the 8 D VGPRs gives half-sums per lane group; xor-16 completes it.
__device__ __forceinline__ float wave_sum_wmma(float x) {
  v2f a; a.x = x;    a.y = 0.0f;
  v2f b; b.x = 1.0f; b.y = 1.0f;
  v8f c = {};
  v8f d = __builtin_amdgcn_wmma_f32_16x16x4_f32(
      false, a, false, b, (short)0, c, false, false);
  float t = d[0] + d[1] + d[2] + d[3] + d[4] + d[5] + d[6] + d[7];
  t += __shfl_xor(t, 16, 32);
  return t;  // every lane holds the full 32-lane sum
}

#if HAVE_TDM
// TDM: DMA a [rows x 256] f32 tile (contiguous rows, stride 256) from global
// memory at gsrc into LDS at lds_byte_off. D# per CDNA5 ISA ch.8:
//   group0: [1:0]=count=1 | [63:32]=lds_addr | [120:64]=global_addr | type=2
//   group1: data_size=2(4B), tensor_dim0=256, tensor_dim1=rows,
//           tile_dim0=256, tile_dim1=rows, tensor_dim0_stride=256
__device__ __forceinline__ void tdm_load_rows(const float* gsrc,
                                              unsigned lds_byte_off, int rows) {
  unsigned long long ga = (unsigned long long)(const void*)gsrc;
  v4u g0;
  g0[0] = 1u;                                              // count=1, user D#
  g0[1] = lds_byte_off;                                    // lds_addr (bytes)
  g0[2] = (unsigned)(ga & 0xFFFFFFFFull);                  // global_addr[31:0]
  g0[3] = (unsigned)((ga >> 32) & 0x01FFFFFFull) | (2u << 30); // [56:32]|type=2
  v8i g1;
  g1[0] = (int)(2u << 16);                // workgroup_mask=0, data_size=4B
  g1[1] = (int)(256u << 16);              // tensor_dim0 = 256 (bits 79:48)
  g1[2] = (int)((unsigned)rows << 16);    // tensor_dim1 = rows (bits 111:80)
  g1[3] = (int)(256u << 16);              // tile_dim0 = 256 (bits 127:112)
  g1[4] = rows;                           // tile_dim1 = rows (bits 143:128)
  g1[5] = 256;                            // tensor_dim0_stride = 256
  g1[6] = 0;
  g1[7] = 0;
  v4i z4 = {0, 0, 0, 0};
#if defined(__clang_major__) && (__clang_major__ >= 23)
  v8i z8 = {0, 0, 0, 0, 0, 0, 0, 0};
  __builtin_amdgcn_tensor_load_to_lds(g0, g1, z4, z4, z8, 0);
#else
  __builtin_amdgcn_tensor_load_to_lds(g0, g1, z4, z4, 0);
#endif
}
#endif

__global__ void init_counts(int* counts) {
  int t = threadIdx.x;
  if (t < NCLS * NB) counts[t] = 0;
}

// One block per (b, i) row; thread j. f2_in = dist^2 to nearest target==c in
// row (else 1e12f), f2_out = nearest !=c. Matches min_j' fl(pen + (j-j')^2):
// (j-j')^2 <= 65025 is fp32-exact; empty-row min is fl(1e12+0) = 1e12f.
__global__ void rowEDT(const int* __restrict__ target,
                       float* __restrict__ f2_in, float* __restrict__ f2_out,
                       int* __restrict__ counts, int c) {
  int bi = blockIdx.x;           // b*256 + i
  int b = bi >> 8, i = bi & 255;
  int j = threadIdx.x;

  __shared__ unsigned char mask[WW];
  __shared__ int cnt;
  if (j == 0) cnt = 0;

  int t = target[(b * HH + i) * WW + j];
  int m = (t == c) ? 1 : 0;
  mask[j] = (unsigned char)m;
  __syncthreads();

  float bin = 1e12f, bout = 1e12f;
#pragma unroll 4
  for (int jp = 0; jp < WW; ++jp) {
    int dj = j - jp;
    float dj2 = (float)(dj * dj);
    if (mask[jp]) bin = fminf(bin, dj2);
    else          bout = fminf(bout, dj2);
  }

  atomicAdd(&cnt, m);            // LDS int atomic: deterministic
  __syncthreads();
  if (j == 0 && cnt > 0) atomicAdd(&counts[(c - 1) * NB + b], 1);

  int idx = (b * HH + i) * WW + j;
  f2_in[idx]  = bin;
  f2_out[idx] = bout;
}

// One block per (b, 4-row output tile); thread j = column. Column min-plus
// with the reference's exact fp32 add sequence. f2 rows are staged into LDS
// in 16-row chunks by the Tensor Data Mover (wave 0 issues, s_wait_tensorcnt,
// workgroup barrier), then each thread folds 4 outputs per staged row pair.
__global__ void colEDTLoss(const float* __restrict__ pred,
                           const int* __restrict__ target,
                           const float* __restrict__ f2_in,
                           const float* __restrict__ f2_out,
                           const int* __restrict__ counts,
                           float* __restrict__ blockSums, int c) {
  int bi = blockIdx.x;                 // b*(HH/IT) + itile
  int b = bi >> 6;                     // HH/IT = 64
  int i0 = (bi & 63) * IT;
  int j = threadIdx.x;

  const float* fin  = f2_in  + b * IMG;
  const float* fout = f2_out + b * IMG;

  __shared__ __align__(16) float lfin[CH * WW];   // 16 KB
  __shared__ __align__(16) float lfout[CH * WW];  // 16 KB

  float d2i[IT], d2o[IT];
#pragma unroll
  for (int u = 0; u < IT; ++u) { d2i[u] = 3.0e38f; d2o[u] = 3.0e38f; }

  for (int chunk = 0; chunk < HH / CH; ++chunk) {
    int base = chunk * CH;
#if HAVE_TDM
    if (threadIdx.x == 0) {
      tdm_load_rows(fin  + base * WW, (unsigned)(unsigned long long)(void*)lfin,  CH);
      tdm_load_rows(fout + base * WW, (unsigned)(unsigned long long)(void*)lfout, CH);
      __builtin_amdgcn_s_wait_tensorcnt(0);
    }
#else
#pragma unroll
    for (int r = 0; r < CH; ++r) {
      lfin[r * WW + j]  = fin[(base + r) * WW + j];
      lfout[r * WW + j] = fout[(base + r) * WW + j];
    }
#endif
    __syncthreads();
#pragma unroll 4
    for (int r = 0; r < CH; ++r) {
      int ip = base + r;
      float fi = lfin[r * WW + j];
      float fo = lfout[r * WW + j];
#pragma unroll
      for (int u = 0; u < IT; ++u) {
        int di = (i0 + u) - ip;
        float di2 = (float)(di * di);
        d2i[u] = fminf(d2i[u], di2 + fi);
        d2o[u] = fminf(d2o[u], di2 + fo);
      }
    }
    __syncthreads();
  }

  int nonempty = counts[(c - 1) * NB + b];
  float contrib = 0.0f;
#pragma unroll
  for (int u = 0; u < IT; ++u) {
    int i = i0 + u;
    float dist = sqrtf(d2i[u]) + sqrtf(d2o[u]);
    if (nonempty == 0) dist = 0.0f;

    int pix = i * WW + j;
    float x0 = pred[(b * 4 + 0) * IMG + pix];
    float x1 = pred[(b * 4 + 1) * IMG + pix];
    float x2 = pred[(b * 4 + 2) * IMG + pix];
    float x3 = pred[(b * 4 + 3) * IMG + pix];
    float mx = fmaxf(fmaxf(x0, x1), fmaxf(x2, x3));
    float e0 = expf(x0 - mx), e1 = expf(x1 - mx);
    float e2 = expf(x2 - mx), e3 = expf(x3 - mx);
    float denom = e0 + e1 + e2 + e3;
    float ec = (c == 1) ? e1 : ((c == 2) ? e2 : e3);
    float p = ec / denom;

    float tc = (target[(b * HH + i) * WW + j] == c) ? 1.0f : 0.0f;
    contrib += fabsf(p - tc) * dist;
  }

  // block reduction: WMMA wave sums -> LDS -> thread 0 (fixed order)
  float wsumv = wave_sum_wmma(contrib);
  __shared__ float wsum[8];
  int lane = threadIdx.x & 31, wid = threadIdx.x >> 5;
  if (lane == 0) wsum[wid] = wsumv;
  __syncthreads();
  if (threadIdx.x == 0) {
    float s = 0.0f;
#pragma unroll
    for (int k = 0; k < 8; ++k) s += wsum[k];
    blockSums[(c - 1) * NBLK2 + bi] = s;
  }
}

// Single block, 256 threads. Fixed-order strided sums over all block sums,
// WMMA wave reduction, final scale. Deterministic.
__global__ void finalize(const float* __restrict__ blockSums, float* __restrict__ out) {
  int tid = threadIdx.x;
  float s = 0.0f;
#pragma unroll
  for (int k = 0; k < NSUMS / 256; ++k)   // 12 strided reads, fixed order
    s += blockSums[k * 256 + tid];

  float wsumv = wave_sum_wmma(s);
  __shared__ float wsum[8];
  int lane = tid & 31, wid = tid >> 5;
  if (lane == 0) wsum[wid] = wsumv;
  __syncthreads();
  if (tid == 0) {
    float g = 0.0f;
#pragma unroll
    for (int k = 0; k < 8; ++k) g += wsum[k];
    out[0] = g / (3.0f * 1048576.0f);     // /(B*H*W) per class, /3 classes
  }
}

extern "C" void kernel_launch(void* const* d_in, const int* in_sizes, int n_in,
                              void* d_out, int out_size, void* d_ws, size_t ws_size,
                              hipStream_t stream) {
  const float* pred   = (const float*)d_in[0];  // [16,4,256,256]
  const int*   target = (const int*)d_in[1];    // [16,256,256]
  float* out = (float*)d_out;

  char* ws = (char*)d_ws;
  int*   counts    = (int*)(ws);                // 48 ints
  float* blockSums = (float*)(ws + 1024);       // NSUMS floats
  float* f2_in     = (float*)(ws + 65536);      // 16*65536 floats (4 MB)
  float* f2_out    = f2_in + NB * IMG;          // 16*65536 floats (4 MB)
  (void)in_sizes; (void)n_in; (void)out_size; (void)ws_size;

  init_counts<<<1, 64, 0, stream>>>(counts);

  for (int c = 1; c <= NCLS; ++c) {
    rowEDT<<<NB * HH, WW, 0, stream>>>(target, f2_in, f2_out, counts, c);
    colEDTLoss<<<NBLK2, WW, 0, stream>>>(pred, target, f2_in, f2_out,
                                         counts, blockSums, c);
  }
  finalize<<<1, 256, 0, stream>>>(blockSums, out);
}